// graph_14594298872375
// MI455X (gfx1250) — compile-verified
//
#include <hip/hip_runtime.h>
#include <cstdint>
#include <cstddef>

// out[c, i] = sum_{e: iInd[e]==i} W[i]^2 * x[c, jInd[e]]
// Pipeline: x[32,N] --LDS-tile-transpose--> xt[N,32]  (one node = one 128B line)
//           edge_scatter: wave32 lane = channel; async-LDS staged indices;
//                         coalesced 128B gather + coalesced 128B fp32 atomic add
//           outt[N,32] --LDS-tile-transpose--> out[32,N]

__global__ void zero4_kernel(float4* __restrict__ p, int n4) {
    int i = blockIdx.x * blockDim.x + threadIdx.x;
    if (i < n4) p[i] = make_float4(0.f, 0.f, 0.f, 0.f);
}

__global__ void zero_kernel(float* __restrict__ p, int n) {
    int i = blockIdx.x * blockDim.x + threadIdx.x;
    if (i < n) p[i] = 0.0f;
}

// src: [32, N] row-major -> dst: [N, 32] row-major. LDS 32x33 tile, conflict-free.
__global__ void transpose_CN_to_NC(const float* __restrict__ src, float* __restrict__ dst, int N) {
    __shared__ float tile[32][33];
    const int tx = threadIdx.x, ty = threadIdx.y;   // blockDim = (32, 8)
    const int n0 = blockIdx.x * 32;
#pragma unroll
    for (int r = 0; r < 32; r += 8) {
        const int c = ty + r;
        const int n = n0 + tx;
        tile[c][tx] = (n < N) ? src[(size_t)c * N + n] : 0.0f;   // coalesced along n
    }
    __syncthreads();
#pragma unroll
    for (int r = 0; r < 32; r += 8) {
        const int n = n0 + ty + r;
        if (n < N) dst[(size_t)n * 32 + tx] = tile[tx][ty + r];  // coalesced along c
    }
}

// src: [N, 32] -> dst: [32, N]
__global__ void transpose_NC_to_CN(const float* __restrict__ src, float* __restrict__ dst, int N) {
    __shared__ float tile[32][33];
    const int tx = threadIdx.x, ty = threadIdx.y;   // blockDim = (32, 8)
    const int n0 = blockIdx.x * 32;
#pragma unroll
    for (int r = 0; r < 32; r += 8) {
        const int n = n0 + ty + r;
        tile[ty + r][tx] = (n < N) ? src[(size_t)n * 32 + tx] : 0.0f;  // coalesced along c
    }
    __syncthreads();
#pragma unroll
    for (int r = 0; r < 32; r += 8) {
        const int c = ty + r;
        const int n = n0 + tx;
        if (n < N) dst[(size_t)c * N + n] = tile[tx][c];               // coalesced along n
    }
}

// One wave (32 lanes) processes a chunk of 32 edges.
//  - 32 (iInd, jInd) pairs staged into LDS via CDNA5 async global->LDS loads
//    (ASYNCcnt path, CDNA5 ISA), overlapped with a
//    global_prefetch of the next chunk's index stream.
//  - Per edge: i, j are wave-uniform -> readfirstlane to SGPRs so W[i] is a
//    scalar load and gather/atomic use uniform-base + lane-offset addressing.
//  - lane = channel: one 128B coalesced gather + one 128B coalesced fp32
//    atomic-add burst per edge (working set ~26MB -> L2-resident at 192MB).
__global__ void edge_scatter(const float* __restrict__ xt,
                             float* __restrict__ outt,
                             const float* __restrict__ W,
                             const int* __restrict__ iInd,
                             const int* __restrict__ jInd,
                             int nEdges) {
    __shared__ int sIdx[8][2][32];                  // per-wave staging: [wave][i/j][edge]
    const int lane = threadIdx.x & 31;
    const int wave = threadIdx.x >> 5;
    const int wavesPerBlock = blockDim.x >> 5;
    const long long gWave  = (long long)blockIdx.x * wavesPerBlock + wave;
    const long long stride = (long long)gridDim.x * wavesPerBlock * 32;

    for (long long e0 = gWave * 32; e0 < nEdges; e0 += stride) {
        long long eL = e0 + lane;
        if (eL >= nEdges) eL = nEdges - 1;          // clamp: duplicate load, masked by kmax
        const unsigned ldsI = (unsigned)(size_t)&sIdx[wave][0][lane];
        const unsigned ldsJ = (unsigned)(size_t)&sIdx[wave][1][lane];
        const int* gI = iInd + eL;
        const int* gJ = jInd + eL;
        // CDNA5 async global->LDS copy (GV mode), tracked by ASYNCcnt.
        asm volatile("global_load_async_to_lds_b32 %0, %1, off"
                     :: "v"(ldsI), "v"(gI) : "memory");
        asm volatile("global_load_async_to_lds_b32 %0, %1, off"
                     :: "v"(ldsJ), "v"(gJ) : "memory");
        if (e0 + stride + lane < nEdges) {          // warm next chunk of the index stream
            __builtin_prefetch(iInd + e0 + stride + lane, 0, 0);
            __builtin_prefetch(jInd + e0 + stride + lane, 0, 0);
        }
        asm volatile("s_wait_asynccnt 0" ::: "memory");

        const long long rem = (long long)nEdges - e0;
        const int kmax = rem >= 32 ? 32 : (int)rem;
        for (int k = 0; k < kmax; ++k) {
            // i, j are identical across the wave: force them into SGPRs so the
            // backend emits scalar W load + SADDR-form vector gather/atomic.
            const int   i  = __builtin_amdgcn_readfirstlane(sIdx[wave][0][k]);
            const int   j  = __builtin_amdgcn_readfirstlane(sIdx[wave][1][k]);
            const float w  = W[i];                  // uniform -> scalar load path
            const float w2 = w * w;
            const float* __restrict__ srcRow = xt   + (size_t)j * 32;  // uniform base
            float*       __restrict__ dstRow = outt + (size_t)i * 32;  // uniform base
            const float v = w2 * srcRow[lane];      // coalesced 128B gather (saddr form)
            unsafeAtomicAdd(&dstRow[lane], v);      // coalesced 128B fadd  (saddr form)
        }
    }
}

// Fallback if workspace is too small: direct atomics into [C, N] layout.
__global__ void edge_direct(const float* __restrict__ x, float* __restrict__ out,
                            const float* __restrict__ W,
                            const int* __restrict__ iInd, const int* __restrict__ jInd,
                            int N, int nEdges) {
    long long e = (long long)blockIdx.x * blockDim.x + threadIdx.x;
    if (e >= nEdges) return;
    const int i = iInd[e], j = jInd[e];
    const float w = W[i];
    const float w2 = w * w;
#pragma unroll
    for (int c = 0; c < 32; ++c)
        unsafeAtomicAdd(&out[(size_t)c * N + i], w2 * x[(size_t)c * N + j]);
}

extern "C" void kernel_launch(void* const* d_in, const int* in_sizes, int n_in,
                              void* d_out, int out_size, void* d_ws, size_t ws_size,
                              hipStream_t stream) {
    const float* x    = (const float*)d_in[0];   // [1, 32, N]
    const float* W    = (const float*)d_in[1];   // [N]
    const int*   iInd = (const int*)d_in[2];     // [E]
    const int*   jInd = (const int*)d_in[3];     // [E]
    const int N = in_sizes[1];
    const int E = in_sizes[2];
    float* out = (float*)d_out;                  // [1, 32, N]

    const size_t needWs = 2ull * (size_t)N * 32 * sizeof(float);  // xt + outt (~12.8 MB)
    if (ws_size >= needWs) {
        float* xt   = (float*)d_ws;
        float* outt = xt + (size_t)N * 32;
        const int nElems = N * 32;               // divisible by 4 (N even)
        const int n4 = nElems / 4;

        zero4_kernel<<<(n4 + 255) / 256, 256, 0, stream>>>((float4*)outt, n4);

        dim3 tb(32, 8);
        const int nTiles = (N + 31) / 32;
        transpose_CN_to_NC<<<nTiles, tb, 0, stream>>>(x, xt, N);

        const long long waves  = ((long long)E + 31) / 32;
        const int       blocks = (int)((waves + 7) / 8);      // 8 waves per 256-thread block
        edge_scatter<<<blocks, 256, 0, stream>>>(xt, outt, W, iInd, jInd, E);

        transpose_NC_to_CN<<<nTiles, tb, 0, stream>>>(outt, out, N);
    } else {
        zero_kernel<<<(out_size + 255) / 256, 256, 0, stream>>>(out, out_size);
        edge_direct<<<(E + 255) / 256, 256, 0, stream>>>(x, out, W, iInd, jInd, N, E);
    }
}